// SparseMultiheadAttention_17506286699026
// MI455X (gfx1250) — compile-verified
//
#include <hip/hip_runtime.h>
#include <hip/hip_bf16.h>

typedef __attribute__((ext_vector_type(2))) float v2f;
typedef __attribute__((ext_vector_type(8))) float v8f;
typedef __attribute__((ext_vector_type(4))) int   v4i;

typedef __attribute__((address_space(1))) v4i as1_v4i;  // global int4
typedef __attribute__((address_space(3))) v4i as3_v4i;  // LDS int4

#define Lnodes 16384
#define Edim   512
#define Hheads 8
#define Dhead  64
#define DEG    16
#define NNZ    (Lnodes * DEG)

// ---------------------------------------------------------------------------
// Async global->LDS copy (CDNA5: GLOBAL_LOAD_ASYNC_TO_LDS_B128, ASYNCcnt).
// Builtin signature (from probe diagnostic): param0 = int4 addrspace(1)*,
// non-const. Guarded: falls back to a register-staged copy if absent.
// ---------------------------------------------------------------------------
#if __has_builtin(__builtin_amdgcn_global_load_async_to_lds_b128)
#define HAS_ASYNC_LDS 1
#else
#define HAS_ASYNC_LDS 0
#endif

__device__ __forceinline__ void copy16_to_lds(const float* __restrict__ g, float* l) {
#if HAS_ASYNC_LDS
    __builtin_amdgcn_global_load_async_to_lds_b128(
        (as1_v4i*)g, (as3_v4i*)l, 0, 0);
#else
    *(float4*)l = *(const float4*)g;
#endif
}

__device__ __forceinline__ void copy_fence() {
#if HAS_ASYNC_LDS
    asm volatile("s_wait_asynccnt 0" ::: "memory");
#endif
    __syncthreads();
}

// ---------------------------------------------------------------------------
// GEMM: OUT[16384][512] = X @ W^T + bias, then * scale.  (torch Linear: W is
// (512x512) row-major, so B[k][n] = W[n][k] -> rows of W are contiguous in k.)
// Block tile 128(M) x 64(N), 256 threads = 8 waves in a 4x2 grid; each wave
// owns a 2x2 set of 16x16 WMMA tiles (4 v8f accumulators).
// Double-buffered LDS, async copies overlap the WMMA loop.
// WMMA: V_WMMA_F32_16X16X4_F32 (exact fp32; reference is fp32).
//   A frag (16x4): lane -> m = l&15, k-pair = (l>>4)*2, v2f
//   B frag (4x16): lane -> n = l&15, same k-pair, v2f
//   C/D (16x16):  VGPR r -> row r + (l>>4)*8, col l&15
// LDS row stride 36 floats: 144B rows keep b128 writes 16B-aligned and the
// v2f fragment reads bank-conflict-free (36*m mod 64 all-distinct, m<16).
// ---------------------------------------------------------------------------
#define LDSTR 36

__global__ __launch_bounds__(256)
void gemm_bias_kernel(const float* __restrict__ X,
                      const float* __restrict__ W,
                      const float* __restrict__ bias,
                      float* __restrict__ OUT,
                      float scale) {
    __shared__ float As[2][128 * LDSTR];
    __shared__ float Bs[2][64 * LDSTR];

    const int tid  = threadIdx.x;
    const int wave = tid >> 5;
    const int lane = tid & 31;
    const int m0 = blockIdx.y * 128;
    const int n0 = blockIdx.x * 64;

    const int mt = (wave & 3) * 2;   // M sub-tiles mt, mt+1 (of 8)
    const int nt = (wave >> 2) * 2;  // N sub-tiles nt, nt+1 (of 4)

    // fragment lane mapping (fp32 WMMA)
    const int fm = lane & 15;
    const int kb = (lane >> 4) << 1;
    const int rowHalf = (lane >> 4) * 8;

    v8f c00 = {}, c01 = {}, c10 = {}, c11 = {};

    // stage one 32-wide K chunk of X (128x32) and W (64x32) into buffer `buf`
    auto stage = [&](int buf, int k0) {
        #pragma unroll
        for (int i = 0; i < 4; ++i) {            // 1024 float4 slots for A
            const int s = tid + i * 256;
            const int row = s >> 3, c = (s & 7) * 4;
            copy16_to_lds(X + (size_t)(m0 + row) * Edim + k0 + c,
                          &As[buf][row * LDSTR + c]);
        }
        #pragma unroll
        for (int i = 0; i < 2; ++i) {            // 512 float4 slots for B
            const int s = tid + i * 256;
            const int row = s >> 3, c = (s & 7) * 4;
            copy16_to_lds(W + (size_t)(n0 + row) * Edim + k0 + c,
                          &Bs[buf][row * LDSTR + c]);
        }
    };

    stage(0, 0);
    copy_fence();

    int buf = 0;
    for (int k0 = 0; k0 < Edim; k0 += 32) {
        if (k0 + 32 < Edim) stage(buf ^ 1, k0 + 32);  // prefetch next chunk

        const float* ar0 = &As[buf][(mt * 16 + fm) * LDSTR + kb];
        const float* ar1 = &As[buf][((mt + 1) * 16 + fm) * LDSTR + kb];
        const float* br0 = &Bs[buf][(nt * 16 + fm) * LDSTR + kb];
        const float* br1 = &Bs[buf][((nt + 1) * 16 + fm) * LDSTR + kb];
        #pragma unroll
        for (int kk = 0; kk < 32; kk += 4) {
            v2f a0 = *(const v2f*)(ar0 + kk);
            v2f a1 = *(const v2f*)(ar1 + kk);
            v2f b0 = *(const v2f*)(br0 + kk);
            v2f b1 = *(const v2f*)(br1 + kk);
            c00 = __builtin_amdgcn_wmma_f32_16x16x4_f32(false, a0, false, b0,
                                                        (short)0, c00, false, false);
            c01 = __builtin_amdgcn_wmma_f32_16x16x4_f32(false, a0, false, b1,
                                                        (short)0, c01, false, false);
            c10 = __builtin_amdgcn_wmma_f32_16x16x4_f32(false, a1, false, b0,
                                                        (short)0, c10, false, false);
            c11 = __builtin_amdgcn_wmma_f32_16x16x4_f32(false, a1, false, b1,
                                                        (short)0, c11, false, false);
        }
        copy_fence();
        buf ^= 1;
    }

    const int nc0 = n0 + nt * 16 + fm;
    const int nc1 = nc0 + 16;
    const float bv0 = bias[nc0];
    const float bv1 = bias[nc1];
    #pragma unroll
    for (int r = 0; r < 8; ++r) {
        const int ma = m0 + mt * 16 + r + rowHalf;
        const int mb = ma + 16;
        OUT[(size_t)ma * Edim + nc0] = (c00[r] + bv0) * scale;
        OUT[(size_t)ma * Edim + nc1] = (c01[r] + bv1) * scale;
        OUT[(size_t)mb * Edim + nc0] = (c10[r] + bv0) * scale;
        OUT[(size_t)mb * Edim + nc1] = (c11[r] + bv1) * scale;
    }
}

// ---------------------------------------------------------------------------
// Edge attention: one block per query row r (edges of r are [16r,16r+16)),
// one wave per head. Scores: lane = (edge j = lane>>1, half = lane&1), each
// lane dots 32 dims; pair-sum via shfl_xor(1). Segment softmax = wave-wide
// shfl reductions (every segment has exactly DEG edges -> no empty-segment
// guard needed; scores duplicated per lane pair -> halve the exp-sum).
// q is pre-scaled by 1/sqrt(D) in the projection.
// ---------------------------------------------------------------------------
__global__ __launch_bounds__(256)
void edge_attn_kernel(const float* __restrict__ q,
                      const float* __restrict__ k,
                      const float* __restrict__ v,
                      const long long* __restrict__ col,  // indices[1], int64
                      float* __restrict__ out) {          // (L, 512)
    const int r    = blockIdx.x;
    const int h    = threadIdx.x >> 5;   // head
    const int lane = threadIdx.x & 31;

    // each lane caches the column of edge (lane&15)
    const int cj = (int)col[(size_t)r * DEG + (lane & 15)];

    // ---- scores ----
    const int j    = lane >> 1;
    const int half = lane & 1;
    const int colj = __shfl(cj, j);
    const float4* q4 = (const float4*)(q + (size_t)r * Edim + h * Dhead + half * 32);
    const float4* k4 = (const float4*)(k + (size_t)colj * Edim + h * Dhead + half * 32);
    float s = 0.f;
    #pragma unroll
    for (int dd = 0; dd < 8; ++dd) {
        float4 a = q4[dd], b = k4[dd];
        s += a.x * b.x + a.y * b.y + a.z * b.z + a.w * b.w;
    }
    s += __shfl_xor(s, 1);  // full dot for edge j (duplicated in lane pair)

    // ---- segment softmax over the 16 edges ----
    float mx = s;
    #pragma unroll
    for (int off = 2; off < 32; off <<= 1) mx = fmaxf(mx, __shfl_xor(mx, off));
    const float e = __expf(s - mx);
    float sum = e;
    #pragma unroll
    for (int off = 1; off < 32; off <<= 1) sum += __shfl_xor(sum, off);
    const float attn = e / (0.5f * sum);  // each edge counted twice

    // ---- weighted value aggregation ----
    float acc0 = 0.f, acc1 = 0.f;
    #pragma unroll
    for (int jj = 0; jj < DEG; ++jj) {
        const float w = __shfl(attn, jj * 2);
        const int   c = __shfl(cj, jj);
        const float* vrow = v + (size_t)c * Edim + h * Dhead;
        acc0 += w * vrow[lane];
        acc1 += w * vrow[lane + 32];
    }
    out[(size_t)r * Edim + h * Dhead + lane]      = acc0;
    out[(size_t)r * Edim + h * Dhead + lane + 32] = acc1;
}

// ---------------------------------------------------------------------------
extern "C" void kernel_launch(void* const* d_in, const int* in_sizes, int n_in,
                              void* d_out, int out_size, void* d_ws, size_t ws_size,
                              hipStream_t stream) {
    const float*     query   = (const float*)d_in[0];
    const float*     key     = (const float*)d_in[1];
    const float*     value   = (const float*)d_in[2];
    const long long* indices = (const long long*)d_in[3];  // (2, NNZ) int64
    const float*     in_w    = (const float*)d_in[4];      // (3E, E)
    const float*     in_b    = (const float*)d_in[5];      // (3E,)
    const float*     out_w   = (const float*)d_in[6];      // (E, E)
    const float*     out_b   = (const float*)d_in[7];      // (E,)
    float*           out     = (float*)d_out;

    float* ws   = (float*)d_ws;
    float* qbuf = ws;                               // L*E
    float* kbuf = ws + (size_t)Lnodes * Edim;       // L*E
    float* vbuf = ws + 2 * (size_t)Lnodes * Edim;   // L*E
    float* abuf = qbuf;  // attn output aliases q: block r reads only q row r
                         // before writing row r -> safe, saves 33 MB of ws

    const dim3 gg(Edim / 64, Lnodes / 128);  // (8, 128)
    const float scaling = 0.125f;            // 1/sqrt(D), D = 64

    gemm_bias_kernel<<<gg, 256, 0, stream>>>(query, in_w,                   in_b,            qbuf, scaling);
    gemm_bias_kernel<<<gg, 256, 0, stream>>>(key,   in_w + Edim * Edim,     in_b + Edim,     kbuf, 1.0f);
    gemm_bias_kernel<<<gg, 256, 0, stream>>>(value, in_w + 2 * Edim * Edim, in_b + 2 * Edim, vbuf, 1.0f);

    edge_attn_kernel<<<Lnodes, 256, 0, stream>>>(qbuf, kbuf, vbuf, indices + NNZ, abuf);

    gemm_bias_kernel<<<gg, 256, 0, stream>>>(abuf, out_w, out_b, out, 1.0f);
}